// SSMClassifier_62302795596597
// MI455X (gfx1250) — compile-verified
//
#include <hip/hip_runtime.h>
#include <math.h>

typedef __attribute__((ext_vector_type(2))) float v2f;
typedef __attribute__((ext_vector_type(8))) float v8f;
typedef __attribute__((address_space(3))) float lds_f;

#define BB 16
#define TT 2048
#define DD 1024
#define NN 1024
#define MM (BB*TT)

// ---------------------------------------------------------------------------
// Cayley: X = (I+S)^-1 (I-S), S = skew - skew^T.  One 32-thread wave per
// 32x32 matrix, Gauss-Jordan (A + A^T = 2I -> stable without pivoting).
// ---------------------------------------------------------------------------
__global__ __launch_bounds__(32) void cayley_kernel(
    const float* __restrict__ Lsk, const float* __restrict__ Rsk,
    float* __restrict__ Lm, float* __restrict__ Rm)
{
    __shared__ float Am[32][33];
    __shared__ float Bm[32][33];
    int bid  = blockIdx.x;            // NL*2*32 = 128 blocks
    int l    = bid >> 6;
    int side = (bid >> 5) & 1;
    int k    = bid & 31;
    const float* skew = (side == 0 ? Lsk : Rsk) + ((size_t)l * 32 + k) * 1024;
    float*       outp = (side == 0 ? Lm  : Rm ) + ((size_t)l * 32 + k) * 1024;

    int r = threadIdx.x;
    for (int j = 0; j < 32; ++j) {
        float s = skew[r * 32 + j] - skew[j * 32 + r];
        float d = (r == j) ? 1.0f : 0.0f;
        Am[r][j] = d + s;
        Bm[r][j] = d - s;
    }
    __syncthreads();
    for (int p = 0; p < 32; ++p) {
        float piv = Am[p][p];
        float f = (r == p) ? 0.0f : Am[r][p] / piv;
        __syncthreads();
        for (int j = 0; j < 32; ++j) {
            Am[r][j] = fmaf(-f, Am[p][j], Am[r][j]);
            Bm[r][j] = fmaf(-f, Bm[p][j], Bm[r][j]);
        }
        __syncthreads();
    }
    float inv = 1.0f / Am[r][r];
    for (int j = 0; j < 32; ++j) outp[r * 32 + j] = Bm[r][j] * inv;
}

// ---------------------------------------------------------------------------
// Embedding gather
// ---------------------------------------------------------------------------
__global__ __launch_bounds__(256) void embed_kernel(
    const int* __restrict__ tokens, const float* __restrict__ emb,
    float* __restrict__ x)
{
    int row = blockIdx.x;
    const float* src = emb + (size_t)tokens[row] * DD;
    float* dst = x + (size_t)row * DD;
    for (int d = threadIdx.x; d < DD; d += 256) dst[d] = src[d];
}

// ---------------------------------------------------------------------------
// LayerNorm, one row per block (256 threads).
// ---------------------------------------------------------------------------
__global__ __launch_bounds__(256) void ln_kernel(
    const float* __restrict__ xbase, size_t rstride,
    const float* __restrict__ g, const float* __restrict__ b,
    float* __restrict__ out)
{
    __shared__ float red[256];
    int row = blockIdx.x, tid = threadIdx.x;
    const float* xr = xbase + (size_t)row * rstride;
    float s = 0.f;
    for (int d = tid; d < DD; d += 256) s += xr[d];
    red[tid] = s; __syncthreads();
    for (int o = 128; o > 0; o >>= 1) { if (tid < o) red[tid] += red[tid + o]; __syncthreads(); }
    float mu = red[0] * (1.0f / DD);
    __syncthreads();
    float v = 0.f;
    for (int d = tid; d < DD; d += 256) { float t = xr[d] - mu; v += t * t; }
    red[tid] = v; __syncthreads();
    for (int o = 128; o > 0; o >>= 1) { if (tid < o) red[tid] += red[tid + o]; __syncthreads(); }
    float inv = rsqrtf(red[0] * (1.0f / DD) + 1e-5f);
    float* orow = out + (size_t)row * DD;
    for (int d = tid; d < DD; d += 256)
        orow[d] = (xr[d] - mu) * inv * g[d] + b[d];
}

// ---------------------------------------------------------------------------
// f32 WMMA GEMM with double-buffered ASYNC global->LDS staging.
// C[M,N] = A[M,K] x B[K,N] (+ fused epilogue).
// 256 threads = 8 waves; 64x64 macro-tile; wave = 16x32 (two 16x16 accums).
// Each thread issues one GLOBAL_LOAD_ASYNC_TO_LDS_B128 for A (64x16 tile)
// and one for B (16x64 tile) per k-step; tile k+1 prefetched into the
// alternate buffer while tile k feeds v_wmma_f32_16x16x4_f32.
// mode 0: sigmoid(acc+bias)   mode 1: acc+bias   mode 2: resid+acc+dvec*u
// M%64==0, N%64==0, K%16==0 guaranteed by caller.
// sA padded to 20 floats/row: rows stay 16B-aligned for the B128 LDS write
// and gcd(20,64)=4 keeps 20*r distinct mod 64 for r<16 -> conflict-free.
// ---------------------------------------------------------------------------
__global__ __launch_bounds__(256) void gemm_wmma_kernel(
    const float* __restrict__ A, const float* __restrict__ B,
    float* __restrict__ C, int M, int K, int N,
    const float* __restrict__ bias,
    const float* __restrict__ resid, const float* __restrict__ u,
    const float* __restrict__ dvec, int mode)
{
    __shared__ float sA[2][64][20];
    __shared__ float sB[2][16][64];

    int tid  = threadIdx.x;
    int lane = tid & 31;
    int wave = tid >> 5;
    int half = lane >> 4;
    int mr   = lane & 15;
    int wm   = wave & 3;     // row subtile 0..3
    int wn   = wave >> 2;    // col half 0..1
    int row0 = blockIdx.y * 64;
    int col0 = blockIdx.x * 64;

    // per-thread async-copy assignments (16B each)
    int ra = tid >> 2;             // A tile row 0..63
    int ca = (tid & 3) * 4;        // A tile col 0,4,8,12
    int rb = tid >> 4;             // B tile row 0..15
    int cb = (tid & 15) * 4;       // B tile col 0..60

    v8f acc0 = {};
    v8f acc1 = {};

#define ISSUE_TILE(bufi, kbase)                                              \
    do {                                                                     \
        unsigned la = (unsigned)(uintptr_t)(lds_f*)&sA[(bufi)][ra][ca];      \
        unsigned lb = (unsigned)(uintptr_t)(lds_f*)&sB[(bufi)][rb][cb];      \
        uint64_t ga = (uint64_t)(uintptr_t)(A + (size_t)(row0 + ra) * K + (kbase) + ca); \
        uint64_t gb = (uint64_t)(uintptr_t)(B + (size_t)((kbase) + rb) * N + col0 + cb); \
        asm volatile("global_load_async_to_lds_b128 %0, %1, off"             \
                     :: "v"(la), "v"(ga) : "memory");                        \
        asm volatile("global_load_async_to_lds_b128 %0, %1, off"             \
                     :: "v"(lb), "v"(gb) : "memory");                        \
    } while (0)

    ISSUE_TILE(0, 0);
    int buf = 0;
    for (int k0 = 0; k0 < K; k0 += 16) {
        asm volatile("s_wait_asynccnt 0" ::: "memory");
        __syncthreads();                     // current tile fully in LDS
        if (k0 + 16 < K) ISSUE_TILE(buf ^ 1, k0 + 16);  // prefetch next
#pragma unroll
        for (int kk = 0; kk < 16; kk += 4) {
            int ks = kk + half * 2;
            v2f a, b0, b1;
            a[0]  = sA[buf][wm * 16 + mr][ks + 0];
            a[1]  = sA[buf][wm * 16 + mr][ks + 1];
            b0[0] = sB[buf][ks + 0][wn * 32 + mr];
            b0[1] = sB[buf][ks + 1][wn * 32 + mr];
            b1[0] = sB[buf][ks + 0][wn * 32 + 16 + mr];
            b1[1] = sB[buf][ks + 1][wn * 32 + 16 + mr];
            acc0 = __builtin_amdgcn_wmma_f32_16x16x4_f32(
                false, a, false, b0, (short)0, acc0, false, false);
            acc1 = __builtin_amdgcn_wmma_f32_16x16x4_f32(
                false, a, false, b1, (short)0, acc1, false, false);
        }
        buf ^= 1;
    }
#undef ISSUE_TILE

    int crow  = row0 + wm * 16 + half * 8;
    int ccol0 = col0 + wn * 32 + mr;
    int ccol1 = ccol0 + 16;
#pragma unroll
    for (int v = 0; v < 8; ++v) {
        int rr = crow + v;
        size_t o0 = (size_t)rr * N + ccol0;
        size_t o1 = (size_t)rr * N + ccol1;
        float x0 = acc0[v], x1 = acc1[v];
        if (mode == 0) {
            x0 = 1.0f / (1.0f + expf(-(x0 + bias[ccol0])));
            x1 = 1.0f / (1.0f + expf(-(x1 + bias[ccol1])));
        } else if (mode == 1) {
            x0 += bias[ccol0];
            x1 += bias[ccol1];
        } else {
            x0 = resid[o0] + x0 + dvec[ccol0] * u[o0];
            x1 = resid[o1] + x1 + dvec[ccol1] * u[o1];
        }
        C[o0] = x0;
        C[o1] = x1;
    }
}

// ---------------------------------------------------------------------------
// Sequential scan. One block per batch element, 1024 threads, state in LDS.
// Thread n = k*32+i owns state element n; Rm row (k,i) and Lm row (i,k)
// pre-loaded into registers (both permutes are the 32x32 transpose, which is
// its own inverse). s_t padded (stride 33) -> conflict-free transpose.
// ---------------------------------------------------------------------------
__global__ __launch_bounds__(1024) void scan_kernel(
    const float* __restrict__ gates, const float* __restrict__ Bu,
    const float* __restrict__ Rm, const float* __restrict__ Lm,
    float* __restrict__ hs)
{
    __shared__ float s_h[NN];
    __shared__ float s_t[32 * 33];

    int a = blockIdx.x;
    int n = threadIdx.x;
    int k = n >> 5;
    int i = n & 31;

    float rR[32], rL[32];
    const float* Rrow = Rm + ((size_t)(k * 32 + i)) * 32;  // Rm[k][i][:]
    const float* Lrow = Lm + ((size_t)(i * 32 + k)) * 32;  // Lm[i][k][:]
#pragma unroll
    for (int j = 0; j < 32; ++j) { rR[j] = Rrow[j]; rL[j] = Lrow[j]; }

    s_h[n] = 0.0f;
    __syncthreads();

    size_t rowBase = (size_t)a * TT;
    for (int t = 0; t < TT; ++t) {
        size_t row = rowBase + t;
        float bu      = Bu[row * NN + n];
        float beta_k  = gates[row * 64 + 32 + k];   // R-step gate (block k)
        float alpha_i = gates[row * 64 + i];        // L-step gate (block i)

        // hr(k,i) = beta_k * sum_j Rm[k,i,j] * h[k*32+j]; store transposed
        float acc = 0.0f;
#pragma unroll
        for (int j = 0; j < 32; ++j) acc = fmaf(rR[j], s_h[(k << 5) + j], acc);
        s_t[i * 33 + k] = acc * beta_k;   // hp[i*32+k] = hr[k*32+i]
        __syncthreads();

        // h_new[n] = alpha_i * sum_j Lm[i,k,j] * hp[i*32+j] + bu[n]
        float acc2 = 0.0f;
#pragma unroll
        for (int j = 0; j < 32; ++j) acc2 = fmaf(rL[j], s_t[i * 33 + j], acc2);
        float hn = fmaf(acc2, alpha_i, bu);
        s_h[n] = hn;
        hs[row * NN + n] = hn;
        __syncthreads();
    }
}

// ---------------------------------------------------------------------------
// Head: gelu(h0 @ h1_w + h1_b) then @ h2_w + h2_b  (tiny: 16 rows)
// ---------------------------------------------------------------------------
__global__ __launch_bounds__(256) void head1_kernel(
    const float* __restrict__ h0, const float* __restrict__ W,
    const float* __restrict__ b, float* __restrict__ h1)
{
    int idx = blockIdx.x * 256 + threadIdx.x;   // 16*2048 outputs
    int row = idx >> 11;
    int col = idx & 2047;
    const float* hr = h0 + (size_t)row * DD;
    float acc = b[col];
    for (int j = 0; j < DD; ++j) acc = fmaf(hr[j], W[(size_t)j * 2048 + col], acc);
    h1[idx] = 0.5f * acc * (1.0f + erff(acc * 0.70710678118654752f));
}

__global__ __launch_bounds__(256) void head2_kernel(
    const float* __restrict__ h1, const float* __restrict__ W,
    const float* __restrict__ b, float* __restrict__ out)
{
    int idx = threadIdx.x;          // 16*16 outputs
    int row = idx >> 4;
    int col = idx & 15;
    const float* hr = h1 + (size_t)row * 2048;
    float acc = b[col];
    for (int j = 0; j < 2048; ++j) acc = fmaf(hr[j], W[(size_t)j * 16 + col], acc);
    out[idx] = acc;
}

// ---------------------------------------------------------------------------
extern "C" void kernel_launch(void* const* d_in, const int* in_sizes, int n_in,
                              void* d_out, int out_size, void* d_ws, size_t ws_size,
                              hipStream_t stream)
{
    (void)in_sizes; (void)n_in; (void)out_size; (void)ws_size;
    const int*   tokens = (const int*)  d_in[0];
    const float* emb    = (const float*)d_in[1];
    const float* L_skew = (const float*)d_in[2];
    const float* R_skew = (const float*)d_in[3];
    const float* Wg_w   = (const float*)d_in[4];
    const float* Wg_b   = (const float*)d_in[5];
    const float* WB_w   = (const float*)d_in[6];
    const float* WB_b   = (const float*)d_in[7];
    const float* C_w    = (const float*)d_in[8];
    const float* D_vec  = (const float*)d_in[9];
    const float* ln_g   = (const float*)d_in[10];
    const float* ln_b   = (const float*)d_in[11];
    const float* hln_g  = (const float*)d_in[12];
    const float* hln_b  = (const float*)d_in[13];
    const float* h1_w   = (const float*)d_in[14];
    const float* h1_b   = (const float*)d_in[15];
    const float* h2_w   = (const float*)d_in[16];
    const float* h2_b   = (const float*)d_in[17];
    float* out = (float*)d_out;

    float* ws = (float*)d_ws;
    size_t off = 0;
    float* x    = ws + off; off += (size_t)MM * DD;
    float* xn   = ws + off; off += (size_t)MM * DD;
    float* BuB  = ws + off; off += (size_t)MM * NN;
    float* hsB  = ws + off; off += (size_t)MM * NN;
    float* gbuf = ws + off; off += (size_t)MM * 64;
    float* LmB  = ws + off; off += (size_t)2 * 32 * 32 * 32;
    float* RmB  = ws + off; off += (size_t)2 * 32 * 32 * 32;
    float* h0B  = ws + off; off += 16 * DD;
    float* h1B  = ws + off; off += 16 * 2048;

    cayley_kernel<<<128, 32, 0, stream>>>(L_skew, R_skew, LmB, RmB);
    embed_kernel<<<MM, 256, 0, stream>>>(tokens, emb, x);

    for (int l = 0; l < 2; ++l) {
        ln_kernel<<<MM, 256, 0, stream>>>(x, (size_t)DD,
                                          ln_g + (size_t)l * DD,
                                          ln_b + (size_t)l * DD, xn);
        // gates = sigmoid(xn @ Wg_w + Wg_b)      M x 1024 x 64
        gemm_wmma_kernel<<<dim3(64 / 64, MM / 64), 256, 0, stream>>>(
            xn, Wg_w + (size_t)l * DD * 64, gbuf, MM, DD, 64,
            Wg_b + (size_t)l * 64, nullptr, nullptr, nullptr, 0);
        // Bu = xn @ WB_w + WB_b                  M x 1024 x 1024
        gemm_wmma_kernel<<<dim3(NN / 64, MM / 64), 256, 0, stream>>>(
            xn, WB_w + (size_t)l * DD * NN, BuB, MM, DD, NN,
            WB_b + (size_t)l * NN, nullptr, nullptr, nullptr, 1);
        // sequential monarch scan -> hs
        scan_kernel<<<BB, 1024, 0, stream>>>(
            gbuf, BuB, RmB + (size_t)l * 32768, LmB + (size_t)l * 32768, hsB);
        // x = x + hs @ C_w + D_vec * xn          M x 1024 x 1024 (in place)
        gemm_wmma_kernel<<<dim3(DD / 64, MM / 64), 256, 0, stream>>>(
            hsB, C_w + (size_t)l * NN * DD, x, MM, NN, DD,
            nullptr, x, xn, D_vec + (size_t)l * DD, 2);
    }

    // head: layernorm of last token rows (stride T*D between batch rows)
    ln_kernel<<<BB, 256, 0, stream>>>(x + (size_t)(TT - 1) * DD, (size_t)TT * DD,
                                      hln_g, hln_b, h0B);
    head1_kernel<<<(16 * 2048) / 256, 256, 0, stream>>>(h0B, h1_w, h1_b, h1B);
    head2_kernel<<<1, 256, 0, stream>>>(h1B, h2_w, h2_b, out);
}